// MultiBoxLoss_9869834846235
// MI455X (gfx1250) — compile-verified
//
#include <hip/hip_runtime.h>
#include <stdint.h>

// Problem constants (from reference)
#define BATCH     64
#define DBOX      8732
#define KCLS      81
#define TILE_ROWS 128          // rows (boxes) per workgroup tile; B*D % 128 == 0 (558848/128 = 4366)
#define T1        256          // threads in kernel 1 (8 waves x 16 rows)
#define NROWS     (BATCH * DBOX)   // 558848
#define NTILES    (NROWS / TILE_ROWS) // 4366

typedef __attribute__((ext_vector_type(2))) float v2f;
typedef __attribute__((ext_vector_type(8))) float v8f;
typedef int v4i_ __attribute__((vector_size(16)));

#if defined(__has_builtin)
#if __has_builtin(__builtin_amdgcn_wmma_f32_16x16x4_f32)
#define HAVE_WMMA_F32X4 1
#endif
#if __has_builtin(__builtin_amdgcn_global_load_async_to_lds_b128)
#define HAVE_ASYNC_LDS 1
#endif
#if __has_builtin(__builtin_amdgcn_s_wait_asynccnt)
#define HAVE_WAIT_ASYNC 1
#endif
#endif

// ---------------------------------------------------------------------------
// Kernel 1: per-box CE (row max -> sum exp -> lse -> ce), positive count,
//           smooth-L1 loc loss. One workgroup = one 128-row tile of conf_preds,
//           staged global->LDS with async-to-LDS copies, exp row-sums via WMMA.
// ---------------------------------------------------------------------------
__global__ __launch_bounds__(T1) void ce_loc_kernel(
    const float* __restrict__ conf, const int* __restrict__ tgt,
    const float* __restrict__ locp, const float* __restrict__ loct,
    float* __restrict__ ce_out, float* __restrict__ locPart,
    unsigned* __restrict__ posPart)
{
    __shared__ __align__(16) float tile[TILE_ROWS * KCLS]; // 41472 B
    __shared__ float    Sbuf[(T1 / 32) * 16];
    __shared__ float    redF[T1];
    __shared__ unsigned redU[T1];

    const long rowBase = (long)blockIdx.x * TILE_ROWS;
    const float* src = conf + rowBase * (long)KCLS;

    // ---- stage tile: 41472 bytes == 2592 x 16B, 16B aligned (41472*blockIdx) ----
#if HAVE_ASYNC_LDS
    for (int i = threadIdx.x; i < (TILE_ROWS * KCLS) / 4; i += T1) {
        __builtin_amdgcn_global_load_async_to_lds_b128(
            (__attribute__((address_space(1))) v4i_*)(src + i * 4),
            (__attribute__((address_space(3))) v4i_*)(&tile[i * 4]),
            0, 0);
    }
#if HAVE_WAIT_ASYNC
    __builtin_amdgcn_s_wait_asynccnt(0);
#else
    asm volatile("s_wait_asynccnt 0" ::: "memory");
#endif
#else
    for (int i = threadIdx.x; i < TILE_ROWS * KCLS; i += T1) tile[i] = src[i];
#endif
    __syncthreads();

    const int lane = threadIdx.x & 31;
    const int wave = threadIdx.x >> 5;
    const int m    = lane & 15;   // row within wave's 16-row group
    const int half = lane >> 4;   // 0 or 1 (which K-pair this lane feeds)
    const int lrow = wave * 16 + m;
    const float* rowp = &tile[lrow * KCLS];

    // ---- row max (two lanes per row; overlapping col 40 is harmless for max) ----
    float mx = -3.4e38f;
    for (int i = 0; i < 41; i++) mx = fmaxf(mx, rowp[half * 40 + i]);
    mx = fmaxf(mx, __shfl_xor(mx, 16));

    // ---- sum of exp(x - rowmax) over K=81 ----
    float S;
#if HAVE_WMMA_F32X4
    {
        v8f acc = {0.f, 0.f, 0.f, 0.f, 0.f, 0.f, 0.f, 0.f};
        v2f bone; bone[0] = 1.f; bone[1] = 1.f;      // B = ones (layout-invariant)
        // A layout (f32 16x4): lanes 0-15 hold K=k,k+1; lanes 16-31 hold K=k+2,k+3
        for (int k = 0; k < 80; k += 4) {
            const int c0 = k + 2 * half;
            v2f a;
            a[0] = __expf(rowp[c0]     - mx);
            a[1] = __expf(rowp[c0 + 1] - mx);
            acc = __builtin_amdgcn_wmma_f32_16x16x4_f32(false, a, false, bone,
                                                        (short)0, acc, false, false);
        }
        { // tail: column 80 (pad 81..83 with exp(-inf)=0)
            v2f a;
            a[0] = half ? 0.f : __expf(rowp[80] - mx);
            a[1] = 0.f;
            acc = __builtin_amdgcn_wmma_f32_16x16x4_f32(false, a, false, bone,
                                                        (short)0, acc, false, false);
        }
        // C layout: lane 0 VGPR i = C[M=i][N=0] (rows 0..7); lane 16 -> rows 8..15
        if (lane == 0) {
#pragma unroll
            for (int i = 0; i < 8; i++) Sbuf[wave * 16 + i] = acc[i];
        } else if (lane == 16) {
#pragma unroll
            for (int i = 0; i < 8; i++) Sbuf[wave * 16 + 8 + i] = acc[i];
        }
        __syncthreads();
        S = Sbuf[wave * 16 + m];
    }
#else
    {
        float p = 0.f;
        for (int i = 0; i < 41; i++) {
            const int c = half * 41 + i;
            if (c < KCLS) p += __expf(rowp[c] - mx);
        }
        p += __shfl_xor(p, 16);
        S = p;
    }
#endif

    // ---- epilogue: ce, pos, smooth-L1 (one row per lane<16) ----
    float ll = 0.f;
    unsigned pp = 0u;
    const long grow = rowBase + lrow;
    if (lane < 16) {
        const int y = tgt[grow];
        const float ce = __logf(S) + mx - rowp[y];
        ce_out[grow] = ce;
        if (y > 0) {
            pp = 1u;
            const float* lp = locp + grow * 4;
            const float* lt = loct + grow * 4;
#pragma unroll
            for (int i = 0; i < 4; i++) {
                const float z = fabsf(lp[i] - lt[i]);
                ll += (z < 1.f) ? 0.5f * z * z : z - 0.5f;
            }
        }
    }
    redF[threadIdx.x] = (lane < 16) ? ll : 0.f;
    redU[threadIdx.x] = (lane < 16) ? pp : 0u;
    __syncthreads();
    for (int s = T1 / 2; s > 0; s >>= 1) {
        if (threadIdx.x < s) {
            redF[threadIdx.x] += redF[threadIdx.x + s];
            redU[threadIdx.x] += redU[threadIdx.x + s];
        }
        __syncthreads();
    }
    if (threadIdx.x == 0) {
        locPart[blockIdx.x] = redF[0];
        posPart[blockIdx.x] = redU[0];
    }
}

// ---------------------------------------------------------------------------
// Kernel 2: hard-negative mining per batch row (radix select of N-th largest
//           masked CE, stable tie-break by index), then conf-loss partial sum.
// ---------------------------------------------------------------------------
__global__ __launch_bounds__(1024) void mine_kernel(
    const int* __restrict__ tgt, const float* __restrict__ ce,
    float* __restrict__ confPart)
{
    __shared__ unsigned ukey[DBOX];      // 34928 B: orderable keys
    __shared__ unsigned hist[256];
    __shared__ unsigned waveTot[32];
    __shared__ unsigned scanArr[1024];
    __shared__ float    redF[1024];
    __shared__ unsigned sh_t, sh_prefix;

    const int  t    = threadIdx.x;
    const int  lane = t & 31;
    const int  wid  = t >> 5;
    const long base = (long)blockIdx.x * DBOX;

    // load keys (cl = pos ? 0 : ce), monotone float->uint map, count positives
    unsigned mypos = 0;
    for (int j = t; j < DBOX; j += 1024) {
        const int  y   = tgt[base + j];
        const bool pos = y > 0;
        const float cl = pos ? 0.0f : ce[base + j];
        const unsigned bits = __float_as_uint(cl);
        ukey[j] = (bits & 0x80000000u) ? ~bits : (bits | 0x80000000u);
        mypos += pos ? 1u : 0u;
    }
    scanArr[t] = mypos;
    __syncthreads();
    for (int s = 512; s > 0; s >>= 1) {
        if (t < s) scanArr[t] += scanArr[t + s];
        __syncthreads();
    }
    const unsigned num_pos = scanArr[0];
    unsigned Nneg = 3u * num_pos;
    if (Nneg > (unsigned)(DBOX - 1)) Nneg = (unsigned)(DBOX - 1);

    // ---- radix select: find the Nneg-th largest key (descending) ----
    unsigned prefix = 0u, tc = Nneg;
    if (Nneg > 0u) {
        for (int p = 3; p >= 0; p--) {
            const int shift = p * 8;
            const unsigned pmask = (p == 3) ? 0u : (0xFFFFFFFFu << (shift + 8));
            if (t < 256) hist[t] = 0u;
            __syncthreads();
            for (int j = t; j < DBOX; j += 1024) {
                const unsigned u = ukey[j];
                if ((u & pmask) == prefix) atomicAdd(&hist[(u >> shift) & 255u], 1u);
            }
            __syncthreads();
            if (t == 0) {
                unsigned cum = 0u, sel = 0u;
                for (int bin = 255; bin >= 0; bin--) {
                    const unsigned c = hist[bin];
                    if (cum + c >= tc) { sel = (unsigned)bin; break; }
                    cum += c;
                }
                sh_t = tc - cum;                       // still needed within chosen bin
                sh_prefix = prefix | (sel << shift);
            }
            __syncthreads();
            tc = sh_t; prefix = sh_prefix;
            __syncthreads();
        }
    }
    const unsigned Tkey = prefix;   // threshold key (value of the Nneg-th largest)
    const unsigned need = tc;       // how many elements EQUAL to Tkey are selected

    // ---- stable (index-order) prefix count of equals, contiguous chunks ----
    const int CH = (DBOX + 1023) / 1024;  // 9
    const int j0 = t * CH;
    const int j1 = (j0 + CH < DBOX) ? (j0 + CH) : DBOX;
    unsigned cnt = 0u;
    if (Nneg > 0u)
        for (int j = j0; j < j1; j++) cnt += (ukey[j] == Tkey) ? 1u : 0u;
    unsigned inc = cnt;                            // wave inclusive scan
    for (int s = 1; s < 32; s <<= 1) {
        const unsigned n = __shfl_up(inc, s);
        if (lane >= s) inc += n;
    }
    if (lane == 31) waveTot[wid] = inc;
    __syncthreads();
    if (wid == 0) {
        unsigned w = waveTot[lane];
        for (int s = 1; s < 32; s <<= 1) {
            const unsigned n = __shfl_up(w, s);
            if (lane >= s) w += n;
        }
        waveTot[lane] = w;
    }
    __syncthreads();
    const unsigned excl = inc - cnt + ((wid > 0) ? waveTot[wid - 1] : 0u);

    // ---- selection + conf-loss partial ----
    float lc = 0.f;
    unsigned run = excl;
    for (int j = j0; j < j1; j++) {
        const int  y   = tgt[base + j];
        const bool pos = y > 0;
        bool sel;
        if (Nneg > 0u) {
            const unsigned u = ukey[j];
            const bool eq = (u == Tkey);
            sel = pos || (u > Tkey) || (eq && run < need);
            if (eq) run++;
        } else {
            sel = pos;
        }
        if (sel) lc += ce[base + j];
    }
    redF[t] = lc;
    __syncthreads();
    for (int s = 512; s > 0; s >>= 1) {
        if (t < s) redF[t] += redF[t + s];
        __syncthreads();
    }
    if (t == 0) confPart[blockIdx.x] = redF[0];
}

// ---------------------------------------------------------------------------
// Kernel 3: deterministic final reduction + normalization.
// out = (loc+conf, conf, loc), each divided by num_matched.
// ---------------------------------------------------------------------------
__global__ __launch_bounds__(1024) void finalize_kernel(
    const float* __restrict__ locPart, const unsigned* __restrict__ posPart,
    const float* __restrict__ confPart, float* __restrict__ out,
    int nLoc, int nConf)
{
    __shared__ float    rf[1024];
    __shared__ unsigned ru[1024];
    const int t = threadIdx.x;
    float ls = 0.f; unsigned ps = 0u;
    for (int i = t; i < nLoc; i += 1024) { ls += locPart[i]; ps += posPart[i]; }
    rf[t] = ls; ru[t] = ps;
    __syncthreads();
    for (int s = 512; s > 0; s >>= 1) {
        if (t < s) { rf[t] += rf[t + s]; ru[t] += ru[t + s]; }
        __syncthreads();
    }
    if (t == 0) {
        float cs = 0.f;
        for (int i = 0; i < nConf; i++) cs += confPart[i];
        const float nm = (float)ru[0];
        const float loc = rf[0] / nm;
        const float cf  = cs / nm;
        out[0] = loc + cf;
        out[1] = cf;
        out[2] = loc;
    }
}

// ---------------------------------------------------------------------------
// Host launch. Inputs: loc_preds, loc_targets, conf_preds, conf_targets.
// Workspace: ce[NROWS] f32 | locPart[NTILES] f32 | posPart[NTILES] u32 |
//            confPart[BATCH] f32  (~2.27 MB total, fully rewritten each call)
// ---------------------------------------------------------------------------
extern "C" void kernel_launch(void* const* d_in, const int* in_sizes, int n_in,
                              void* d_out, int out_size, void* d_ws, size_t ws_size,
                              hipStream_t stream)
{
    const float* locp = (const float*)d_in[0];
    const float* loct = (const float*)d_in[1];
    const float* conf = (const float*)d_in[2];
    const int*   tgt  = (const int*)d_in[3];
    float* out = (float*)d_out;

    float*    ce       = (float*)d_ws;
    float*    locPart  = ce + NROWS;
    unsigned* posPart  = (unsigned*)(locPart + NTILES);
    float*    confPart = (float*)(posPart + NTILES);

    ce_loc_kernel<<<NTILES, T1, 0, stream>>>(conf, tgt, locp, loct,
                                             ce, locPart, posPart);
    mine_kernel<<<BATCH, 1024, 0, stream>>>(tgt, ce, confPart);
    finalize_kernel<<<1, 1024, 0, stream>>>(locPart, posPart, confPart, out,
                                            NTILES, BATCH);
}